// PauliXGate_50964081934938
// MI455X (gfx1250) — compile-verified
//
#include <hip/hip_runtime.h>
#include <hip/hip_bf16.h>

// y = x @ tridiag(1 on super/sub diagonal)  ==  y[..,c] = x[..,c-1] + x[..,c+1]
// x: (4, 4096, 4096) f32 -> 16384 independent rows of 4096 floats.
// Memory-bound stencil: 512 MB total HBM traffic, ~22us at 23.3 TB/s.
// Data path: CDNA5 async global->LDS b128 copies (ASYNCcnt), compute from LDS.

#define HID      4096
#define THREADS  256   // 8 wave32s per workgroup

__global__ __launch_bounds__(THREADS) void paulix_stencil_kernel(
    const float* __restrict__ x, float* __restrict__ y) {
  __shared__ __align__(16) float tile[HID];

  const int    tid  = threadIdx.x;
  const size_t base = (size_t)blockIdx.x * HID;

  // ---- Stage the whole 16KB row into LDS with async b128 copies. ----
  // Each thread moves 4 x 16B, strided by 4096B via the instruction
  // immediate offset (ISA: INST_OFFSET is added to BOTH the global memory
  // address and the LDS address), so a single address pair covers the row.
  {
    uint64_t gaddr = (uint64_t)(uintptr_t)(x + base + (size_t)tid * 4);
    uint32_t laddr = (uint32_t)(uintptr_t)(&tile[tid * 4]);  // low 32b = LDS offset
    asm volatile(
        "global_load_async_to_lds_b128 %0, %1, off\n\t"
        "global_load_async_to_lds_b128 %0, %1, off offset:4096\n\t"
        "global_load_async_to_lds_b128 %0, %1, off offset:8192\n\t"
        "global_load_async_to_lds_b128 %0, %1, off offset:12288"
        :
        : "v"(laddr), "v"(gaddr)
        : "memory");
    // Wave-local completion of the async copies, then cross-wave barrier.
    asm volatile("s_wait_asynccnt 0x0" ::: "memory");
  }
  __syncthreads();

  // ---- Shift-add from LDS, 4 outputs per thread per iteration. ----
#pragma unroll
  for (int it = 0; it < 4; ++it) {
    const int c0 = it * (THREADS * 4) + tid * 4;  // 0..4092, multiple of 4

    const float4 c    = *reinterpret_cast<const float4*>(&tile[c0]);
    const float  left  = (c0 == 0)       ? 0.0f : tile[c0 - 1];
    const float  right = (c0 == HID - 4) ? 0.0f : tile[c0 + 4];

    float4 o;
    o.x = left + c.y;   // y[c0]   = x[c0-1] + x[c0+1]   (x[-1] := 0)
    o.y = c.x  + c.z;   // y[c0+1]
    o.z = c.y  + c.w;   // y[c0+2]
    o.w = c.z  + right; // y[c0+3] = x[c0+2] + x[c0+4]   (x[4096] := 0)

    *reinterpret_cast<float4*>(y + base + c0) = o;
  }
}

extern "C" void kernel_launch(void* const* d_in, const int* in_sizes, int n_in,
                              void* d_out, int out_size, void* d_ws, size_t ws_size,
                              hipStream_t stream) {
  const float* x = (const float*)d_in[0];
  float*       y = (float*)d_out;
  const int rows = in_sizes[0] / HID;  // 4 * 4096 = 16384 rows
  paulix_stencil_kernel<<<rows, THREADS, 0, stream>>>(x, y);
}